// LargeKernelAttention_24567212933351
// MI455X (gfx1250) — compile-verified
//
#include <hip/hip_runtime.h>
#include <math.h>

#define B_  8
#define C_  64
#define H_  256
#define W_  256
#define HW_ (H_ * W_)
#define CHW_ (C_ * HW_)
#define TOTAL_ (B_ * CHW_)

typedef float v2f __attribute__((ext_vector_type(2)));
typedef float v8f __attribute__((ext_vector_type(8)));

// ---------------------------------------------------------------------------
// Stage 1: 5x5 depthwise conv, pad 2.  One output element per thread,
// coalesced along W; stencil neighborhood reuse is served by L0/L2
// (whole tensor fits in the 192 MB L2).
// ---------------------------------------------------------------------------
__global__ void __launch_bounds__(256)
dw5x5_kernel(const float* __restrict__ x, const float* __restrict__ lw,
             float* __restrict__ out)
{
    int id = blockIdx.x * 256 + threadIdx.x;
    int w = id % W_;
    int h = (id / W_) % H_;
    int c = (id / HW_) % C_;
    const float* wp = lw + c * 25;
    const float* xp = x + (id - h * W_ - w);   // base of this (b,c) plane

    float acc = 0.0f;
#pragma unroll
    for (int kh = 0; kh < 5; ++kh) {
        int hh = h + kh - 2;
        if ((unsigned)hh < (unsigned)H_) {
#pragma unroll
            for (int kw = 0; kw < 5; ++kw) {
                int ww = w + kw - 2;
                if ((unsigned)ww < (unsigned)W_)
                    acc = fmaf(xp[hh * W_ + ww], wp[kh * 5 + kw], acc);
            }
        }
    }
    out[id] = acc;
}

// ---------------------------------------------------------------------------
// Stage 2: 1x21 depthwise conv (horizontal), pad 10.
// ---------------------------------------------------------------------------
__global__ void __launch_bounds__(256)
dwH_kernel(const float* __restrict__ x, const float* __restrict__ hw,
           float* __restrict__ out)
{
    int id = blockIdx.x * 256 + threadIdx.x;
    int w = id % W_;
    int c = (id / HW_) % C_;
    const float* wp = hw + c * 21;
    const float* rp = x + (id - w);            // base of this row

    float acc = 0.0f;
#pragma unroll
    for (int k = 0; k < 21; ++k) {
        int ww = w + k - 10;
        if ((unsigned)ww < (unsigned)W_)
            acc = fmaf(rp[ww], wp[k], acc);
    }
    out[id] = acc;
}

// ---------------------------------------------------------------------------
// Stage 3: 21x1 depthwise conv (vertical), pad 10.
// ---------------------------------------------------------------------------
__global__ void __launch_bounds__(256)
dwV_kernel(const float* __restrict__ x, const float* __restrict__ vw,
           float* __restrict__ out)
{
    int id = blockIdx.x * 256 + threadIdx.x;
    int w = id % W_;
    int h = (id / W_) % H_;
    int c = (id / HW_) % C_;
    const float* wp = vw + c * 21;
    const float* xp = x + (id - h * W_ - w);   // base of this (b,c) plane

    float acc = 0.0f;
#pragma unroll
    for (int k = 0; k < 21; ++k) {
        int hh = h + k - 10;
        if ((unsigned)hh < (unsigned)H_)
            acc = fmaf(xp[hh * W_ + w], wp[k], acc);
    }
    out[id] = acc;
}

// ---------------------------------------------------------------------------
// Stage 4: 1x1 pointwise (64x64 channel mix) as f32 WMMA GEMM, fused with
// BatchNorm(inference) + sigmoid gate:  out = x * sigmoid(attn*inv + bias).
//
// Per wave32: D[16 outC x 16 pixels] tiles, all 64 outC via 4 v8f
// accumulators; K=64 swept in 16 steps of V_WMMA_F32_16X16X4_F32.
// f32 A fragment layout (ISA 7.12.2): lane<16 -> {K=k0,k0+1}, lane>=16 ->
// {K=k0+2,k0+3}, M = lane&15.  B mirrors with N = lane&15.  D: lane<16 ->
// M=r, lane>=16 -> M=8+r, N = lane&15.
// ---------------------------------------------------------------------------
__global__ void __launch_bounds__(256)
pw_bn_gate_wmma_kernel(const float* __restrict__ attn,   // stage-3 result
                       const float* __restrict__ x,
                       const float* __restrict__ pw,     // 64x64 row-major [out][in]
                       const float* __restrict__ gamma,
                       const float* __restrict__ beta,
                       const float* __restrict__ mean,
                       const float* __restrict__ var,
                       float* __restrict__ out)
{
    __shared__ float Wlds[C_ * C_];   // 16 KB weight tile, loaded once/block
    __shared__ float invS[C_];
    __shared__ float biasS[C_];

    int tid = threadIdx.x;
    for (int i = tid; i < C_ * C_; i += 256) Wlds[i] = pw[i];
    if (tid < C_) {
        float iv = gamma[tid] * rsqrtf(var[tid] + 1e-5f);
        invS[tid]  = iv;
        biasS[tid] = beta[tid] - mean[tid] * iv;
    }
    __syncthreads();

    int wave = tid >> 5;
    int lane = tid & 31;

    // 16 consecutive pixels per wave; HW_ is a multiple of 128 so a wave
    // never straddles a batch image.
    int p0  = (blockIdx.x * 8 + wave) * 16;
    int b   = p0 / HW_;
    int pix = p0 % HW_;
    const float* aBase = attn + b * CHW_ + pix;   // + c*HW_ + n
    const float* xBase = x    + b * CHW_ + pix;
    float*       oBase = out  + b * CHW_ + pix;

    int n     = lane & 15;          // pixel within tile / M row for A
    int khalf = (lane >> 4) * 2;    // lanes 16-31 carry K = k0+2, k0+3

    v8f acc0 = {}, acc1 = {}, acc2 = {}, acc3 = {};

#pragma unroll
    for (int k0 = 0; k0 < C_; k0 += 4) {
        // B fragment: attn channels (k0+khalf .. k0+khalf+1) at pixel n
        v2f bf;
        bf.x = aBase[(k0 + khalf + 0) * HW_ + n];
        bf.y = aBase[(k0 + khalf + 1) * HW_ + n];

        // A fragments: W[m][k], m = mt*16 + (lane&15)
        const float* wr = &Wlds[n * C_ + k0 + khalf];
        v2f a0; a0.x = wr[0 * 16 * C_]; a0.y = wr[0 * 16 * C_ + 1];
        v2f a1; a1.x = wr[1 * 16 * C_]; a1.y = wr[1 * 16 * C_ + 1];
        v2f a2; a2.x = wr[2 * 16 * C_]; a2.y = wr[2 * 16 * C_ + 1];
        v2f a3; a3.x = wr[3 * 16 * C_]; a3.y = wr[3 * 16 * C_ + 1];

        acc0 = __builtin_amdgcn_wmma_f32_16x16x4_f32(false, a0, false, bf,
                                                     (short)0, acc0, false, false);
        acc1 = __builtin_amdgcn_wmma_f32_16x16x4_f32(false, a1, false, bf,
                                                     (short)0, acc1, false, false);
        acc2 = __builtin_amdgcn_wmma_f32_16x16x4_f32(false, a2, false, bf,
                                                     (short)0, acc2, false, false);
        acc3 = __builtin_amdgcn_wmma_f32_16x16x4_f32(false, a3, false, bf,
                                                     (short)0, acc3, false, false);
    }

    // Fused epilogue: BN + sigmoid gate + store.
    int mrow = (lane >> 4) * 8;     // lanes 16-31 hold M = 8..15
    v8f accs[4] = {acc0, acc1, acc2, acc3};
#pragma unroll
    for (int mt = 0; mt < 4; ++mt) {
#pragma unroll
        for (int r = 0; r < 8; ++r) {
            int c   = mt * 16 + mrow + r;
            float v = fmaf(accs[mt][r], invS[c], biasS[c]);
            float sg = 1.0f / (1.0f + __expf(-v));
            int idx = c * HW_ + n;
            oBase[idx] = xBase[idx] * sg;
        }
    }
}

// ---------------------------------------------------------------------------
// Launch: ping-pong between d_ws (128 MiB) and d_out (used as scratch for
// the middle stages, fully overwritten by the final fused kernel).
//   s1: x   -> ws     (5x5 dw)
//   s2: ws  -> d_out  (1x21 dw)
//   s3: d_out -> ws   (21x1 dw)
//   s4: ws, x -> d_out (WMMA pointwise + BN + sigmoid gate)
// ---------------------------------------------------------------------------
extern "C" void kernel_launch(void* const* d_in, const int* in_sizes, int n_in,
                              void* d_out, int out_size, void* d_ws, size_t ws_size,
                              hipStream_t stream) {
    const float* x       = (const float*)d_in[0];
    const float* local_w = (const float*)d_in[1];
    const float* h_w     = (const float*)d_in[2];
    const float* v_w     = (const float*)d_in[3];
    const float* pw_w    = (const float*)d_in[4];
    const float* gamma   = (const float*)d_in[5];
    const float* beta    = (const float*)d_in[6];
    const float* mean    = (const float*)d_in[7];
    const float* var     = (const float*)d_in[8];

    float* out = (float*)d_out;
    float* ws  = (float*)d_ws;          // needs TOTAL_ * 4 = 128 MiB

    dim3 blk(256);
    dim3 grd(TOTAL_ / 256);             // exact: 131072 blocks

    dw5x5_kernel<<<grd, blk, 0, stream>>>(x, local_w, ws);
    dwH_kernel  <<<grd, blk, 0, stream>>>(ws, h_w, out);
    dwV_kernel  <<<grd, blk, 0, stream>>>(out, v_w, ws);

    // 524288 pixels total, 128 pixels per 256-thread block -> 4096 blocks,
    // every wave fully populated (EXEC all-1s, as WMMA requires).
    dim3 ggrd((B_ * HW_) / 128);
    pw_bn_gate_wmma_kernel<<<ggrd, blk, 0, stream>>>(ws, x, pw_w, gamma, beta,
                                                     mean, var, out);
}